// GraphConvLayer_65824668778576
// MI455X (gfx1250) — compile-verified
//
#include <hip/hip_runtime.h>
#include <hip/hip_bf16.h>

// ---------------------------------------------------------------------------
// GraphConv: out = segment_sum(edge_weight * x[row], col) @ W.T + b
//   Phase 1: zero agg scratch (d_ws)
//   Phase 2: edge scatter with native f32 global atomics (memory-bound)
//   Phase 3: f32 WMMA GEMM (V_WMMA_F32_16X16X4_F32), fused bias
// ---------------------------------------------------------------------------

typedef float v2f __attribute__((ext_vector_type(2)));
typedef float v8f __attribute__((ext_vector_type(8)));

#define GC_DIM 128
#define LDS_STRIDE 132   // 132 floats/row: 16B-aligned rows, conflict-free lanes

// --------------------------- Phase 1: zero agg ------------------------------
__global__ __launch_bounds__(256) void gc_zero_kernel(float4* __restrict__ agg4, int n4) {
    int i = blockIdx.x * blockDim.x + threadIdx.x;
    if (i < n4) agg4[i] = make_float4(0.f, 0.f, 0.f, 0.f);
}

// --------------------------- Phase 2: scatter -------------------------------
// One wave32 per edge; lane j handles dims [4j, 4j+3].
__global__ __launch_bounds__(256) void gc_scatter_kernel(const float* __restrict__ x,
                                                         const long long* __restrict__ ei,
                                                         const float* __restrict__ ew,
                                                         float* __restrict__ agg,
                                                         int n_edges) {
    const int gtid = blockIdx.x * blockDim.x + threadIdx.x;
    const int e    = gtid >> 5;
    const int lane = gtid & 31;
    if (e >= n_edges) return;

    const long long src = ei[e];             // row
    const long long dst = ei[n_edges + e];   // col
    const float w = ew[e];

    const float4 v = ((const float4*)(x + src * GC_DIM))[lane];
    float* dp = agg + dst * GC_DIM + lane * 4;
    // native global_atomic_add_f32 (relaxed, device scope)
    __hip_atomic_fetch_add(dp + 0, w * v.x, __ATOMIC_RELAXED, __HIP_MEMORY_SCOPE_AGENT);
    __hip_atomic_fetch_add(dp + 1, w * v.y, __ATOMIC_RELAXED, __HIP_MEMORY_SCOPE_AGENT);
    __hip_atomic_fetch_add(dp + 2, w * v.z, __ATOMIC_RELAXED, __HIP_MEMORY_SCOPE_AGENT);
    __hip_atomic_fetch_add(dp + 3, w * v.w, __ATOMIC_RELAXED, __HIP_MEMORY_SCOPE_AGENT);
}

// ------------------- Phase 3: WMMA GEMM  out = agg @ W.T + b ----------------
// Block = 256 threads = 8 wave32s. Block owns a 16-row M-tile of agg (staged
// to LDS once); wave w owns N-tile [16w, 16w+16). K=128 -> 32 x WMMA f32 k4.
__global__ __launch_bounds__(256) void gc_gemm_bias_kernel(const float* __restrict__ agg,
                                                           const float* __restrict__ Wm,
                                                           const float* __restrict__ bias,
                                                           float* __restrict__ out,
                                                           int n_nodes) {
    __shared__ float lds_a[16 * LDS_STRIDE];

    const int m0 = blockIdx.x * 16;
    const int t  = threadIdx.x;

    // ---- stage 16x128 A-tile: each thread copies 8 contiguous floats ----
    {
        const int r  = t >> 4;         // 0..15  (row in tile)
        const int c0 = (t & 15) * 8;   // 0..120 (col start)
        float4 v0 = make_float4(0.f, 0.f, 0.f, 0.f), v1 = v0;
        if (m0 + r < n_nodes) {
            const float4* g = (const float4*)(agg + (size_t)(m0 + r) * GC_DIM + c0);
            v0 = g[0];
            v1 = g[1];
        }
        float4* s = (float4*)(lds_a + r * LDS_STRIDE + c0);  // 16B aligned
        s[0] = v0;
        s[1] = v1;
    }
    __syncthreads();

    const int wave = t >> 5;      // 0..7 -> N-tile
    const int lane = t & 31;
    const int half = lane >> 4;   // 0: K pair {0,1}; 1: K pair {2,3}
    const int l16  = lane & 15;   // A: M index / B: N index / D: N index
    const int n0   = wave * 16;

    const float* arow = lds_a + l16 * LDS_STRIDE;        // A row M=l16
    const float* brow = Wm + (size_t)(n0 + l16) * GC_DIM; // B col n = W row n

    v8f c = {};
#pragma unroll
    for (int ks = 0; ks < 32; ++ks) {
        const int k = ks * 4 + half * 2;
        v2f a = *(const v2f*)(arow + k);   // ds_load_b64
        v2f bf = *(const v2f*)(brow + k);  // global_load_b64 (L0/L2 resident)
        // D = A(16x4) x B(4x16) + C   -> v_wmma_f32_16x16x4_f32
        c = __builtin_amdgcn_wmma_f32_16x16x4_f32(
                /*neg_a=*/false, a, /*neg_b=*/false, bf,
                /*c_mod=*/(short)0, c, /*reuse_a=*/false, /*reuse_b=*/false);
    }

    // C/D layout: VGPR i -> M = i (lanes 0-15) or M = 8+i (lanes 16-31), N = l16
    const float bn = bias[n0 + l16];
#pragma unroll
    for (int i = 0; i < 8; ++i) {
        const int m = m0 + i + half * 8;
        if (m < n_nodes) out[(size_t)m * GC_DIM + n0 + l16] = c[i] + bn;
    }
}

// ---------------------------------------------------------------------------
extern "C" void kernel_launch(void* const* d_in, const int* in_sizes, int n_in,
                              void* d_out, int out_size, void* d_ws, size_t ws_size,
                              hipStream_t stream) {
    const float*     x  = (const float*)d_in[0];      // [N, 128]
    const long long* ei = (const long long*)d_in[1];  // [2, E] int64
    const float*     ew = (const float*)d_in[2];      // [E]
    const float*     Wm = (const float*)d_in[3];      // [128, 128]
    const float*     b  = (const float*)d_in[4];      // [128]
    float*           out = (float*)d_out;             // [N, 128]
    float*           agg = (float*)d_ws;              // [N, 128] scratch

    const int n_nodes = in_sizes[0] / GC_DIM;
    const int n_edges = in_sizes[2];

    // Phase 1: zero agg
    {
        const int n4 = (n_nodes * GC_DIM) / 4;
        const int blocks = (n4 + 255) / 256;
        gc_zero_kernel<<<blocks, 256, 0, stream>>>((float4*)agg, n4);
    }
    // Phase 2: scatter-add (1 wave per edge)
    {
        const long long threads = (long long)n_edges * 32;
        const int blocks = (int)((threads + 255) / 256);
        gc_scatter_kernel<<<blocks, 256, 0, stream>>>(x, ei, ew, agg, n_edges);
    }
    // Phase 3: WMMA GEMM + bias
    {
        const int blocks = (n_nodes + 15) / 16;
        gc_gemm_bias_kernel<<<blocks, 256, 0, stream>>>(agg, Wm, b, out, n_nodes);
    }
}